// TreeTransformer_89464168776202
// MI455X (gfx1250) — compile-verified
//
#include <hip/hip_runtime.h>

typedef __attribute__((ext_vector_type(2))) float v2f;
typedef __attribute__((ext_vector_type(4))) float v4f;
typedef __attribute__((ext_vector_type(8))) float v8f;

#define BB    4
#define AA    8
#define NN    4096
#define FIN   128
#define HIDW  512
#define MTOT  (BB * AA * NN)      /* 131072 rows */
#define PITCH 130                 /* LDS row pitch (dwords): even -> b64 aligned; 130%64=2 -> conflict-free */
#define MTILES 4                  /* 4 x 16 = 64 rows per block */
#define ROWS  (MTILES * 16)

// ---------------------------------------------------------------------------
// Kernel 1: global max over node_order (single block, deterministic, writes ws)
// ---------------------------------------------------------------------------
__global__ __launch_bounds__(1024)
void node_order_max_kernel(const int* __restrict__ no, int count, int* __restrict__ out) {
    __shared__ int sm[1024];
    int t = threadIdx.x;
    int m = 0;                                  // node_order values are >= 0
    for (int i = t; i < count; i += 1024) {
        int v = no[i];
        m = v > m ? v : m;
    }
    sm[t] = m;
    __syncthreads();
    for (int s = 512; s > 0; s >>= 1) {
        if (t < s) { int a = sm[t], b2 = sm[t + s]; sm[t] = a > b2 ? a : b2; }
        __syncthreads();
    }
    if (t == 0) out[0] = sm[0];
}

// ---------------------------------------------------------------------------
// Kernel 2: fused GEMM (forest @ W + bias) + tree positional encoding epilogue
//   block: 256 threads = 8 waves; owns 64 output rows x all 512 cols
//   wave w: cols [64w, 64w+64); per 16-col tile, W slice (128x16) lives in
//   64 VGPRs and is reused across 4 row tiles (A streamed from LDS)
// ---------------------------------------------------------------------------
__global__ __launch_bounds__(256)
void tree_embed_gemm_kernel(const float* __restrict__ forest,
                            const float* __restrict__ W,
                            const float* __restrict__ bias,
                            const int*   __restrict__ node_order,
                            const int*   __restrict__ maxbuf,
                            float*       __restrict__ out) {
    __shared__ __align__(16) float As[ROWS * PITCH];   // 64 x 130 dwords = 33.3 KB
    __shared__ int s_h[ROWS];   // precomputed one-hot column per row (-1 = none)

    const int tid     = threadIdx.x;
    const int lane    = tid & 31;
    const int wave    = tid >> 5;
    const int rowbase = blockIdx.x * ROWS;

    // ---- stage A tile: 64 rows x 128 floats = contiguous 32KB, NT loads ----
    const v4f* src = (const v4f*)(forest + (size_t)rowbase * FIN);
    #pragma unroll
    for (int it = 0; it < (ROWS * FIN / 4) / 256; ++it) {   // 8 float4 per thread
        int i = tid + it * 256;            // 0..2047 float4s
        v4f v = __builtin_nontemporal_load(&src[i]);
        int r = i >> 5;                    // 32 float4 per row
        int c = (i & 31) << 2;
        float* dst = &As[r * PITCH + c];
        dst[0] = v.x; dst[1] = v.y; dst[2] = v.z; dst[3] = v.w;
    }
    // ---- stage per-row PE target column ----
    if (tid < ROWS) {
        int g  = rowbase + tid;
        int n  = g & (NN - 1);             // node index within tree
        int no = node_order[g];
        int maxd = maxbuf[0];
        bool valid = (n >= 1) && (no < maxd);
        s_h[tid] = valid ? (3 * no + ((n + 2) % 3)) : -1;
    }
    __syncthreads();

    // ISA 32-bit 16x4/4x16 frag layout: lane%16 = M (or N); VGPR0/1 = K pair 2*(lane/16)+{0,1}
    const int m16   = lane & 15;
    const int khalf = (lane >> 4) << 1;
    const int mbase = (lane >> 4) << 3;    // C/D layout: lanes 16-31 hold M = r+8

    #pragma unroll 1
    for (int nt = 0; nt < 4; ++nt) {
        const int col = wave * 64 + nt * 16 + m16;
        const float* Wc = W + col;         // W[k][col], row stride HIDW

        // ---- load full-K W slice for this 16-col tile into registers ----
        v2f bv[32];
        #pragma unroll
        for (int kc = 0; kc < 32; ++kc) {
            const int k = kc * 4 + khalf;
            bv[kc].x = Wc[(size_t)k * HIDW];
            bv[kc].y = Wc[(size_t)(k + 1) * HIDW];
        }
        const float bs = bias[col];

        // ---- reuse W registers across 4 row tiles; A streams from LDS ----
        #pragma unroll 1
        for (int mt = 0; mt < MTILES; ++mt) {
            const float* Arow = &As[(mt * 16 + m16) * PITCH];
            v8f acc = {0.f, 0.f, 0.f, 0.f, 0.f, 0.f, 0.f, 0.f};
            #pragma unroll
            for (int kc = 0; kc < 32; ++kc) {
                v2f a = *(const v2f*)&Arow[kc * 4 + khalf];
                acc = __builtin_amdgcn_wmma_f32_16x16x4_f32(
                    /*neg_a=*/false, a, /*neg_b=*/false, bv[kc],
                    /*c_mod=*/(short)0, acc, /*reuse_a=*/false, /*reuse_b=*/false);
            }
            // ---- epilogue: bias + one-hot PE, non-temporal streaming stores ----
            #pragma unroll
            for (int r = 0; r < 8; ++r) {
                const int mm = mt * 16 + mbase + r;
                float v = acc[r] + bs + ((s_h[mm] == col) ? 1.0f : 0.0f);
                __builtin_nontemporal_store(v, &out[(size_t)(rowbase + mm) * HIDW + col]);
            }
        }
    }
}

// ---------------------------------------------------------------------------
extern "C" void kernel_launch(void* const* d_in, const int* in_sizes, int n_in,
                              void* d_out, int out_size, void* d_ws, size_t ws_size,
                              hipStream_t stream) {
    const float* forest     = (const float*)d_in[0];
    const float* W          = (const float*)d_in[1];
    const float* bias       = (const float*)d_in[2];
    // d_in[3] = adjacency (unused), d_in[5] = edge_order (unused)
    const int*   node_order = (const int*)d_in[4];
    float*       out        = (float*)d_out;
    int*         maxbuf     = (int*)d_ws;

    const int count = in_sizes[4];         // B*A*N = 131072

    node_order_max_kernel<<<1, 1024, 0, stream>>>(node_order, count, maxbuf);
    tree_embed_gemm_kernel<<<MTOT / ROWS, 256, 0, stream>>>(
        forest, W, bias, node_order, maxbuf, out);
}